// PointNetFeaturePropagation_30683246363224
// MI455X (gfx1250) — compile-verified
//
#include <hip/hip_runtime.h>
#include <hip/hip_bf16.h>
#include <math.h>

// Problem constants (match setup_inputs)
#define B_    8
#define N_    8192
#define S_    2048
#define C1_   128
#define C2_   256
#define TDIM_ 128
#define K0_   384
#define M_    256
#define EPS_DIST 1e-8f
#define EPS_BN   1e-5f

#define KT_   16     // K-tile staged per step
#define MB_   32     // M rows per workgroup
#define NB_   256    // N cols per workgroup

typedef float v2f __attribute__((ext_vector_type(2)));
typedef float v8f __attribute__((ext_vector_type(8)));
typedef int   i32x4 __attribute__((vector_size(16)));
typedef int   i32x2 __attribute__((vector_size(8)));

#define AS1 __attribute__((address_space(1)))
#define AS3 __attribute__((address_space(3)))

#if defined(__gfx1250__) &&                                         \
    __has_builtin(__builtin_amdgcn_global_load_async_to_lds_b128) && \
    __has_builtin(__builtin_amdgcn_global_load_async_to_lds_b64) &&  \
    __has_builtin(__builtin_amdgcn_s_wait_asynccnt)
#define USE_ASYNC 1
#else
#define USE_ASYNC 0
#endif

__device__ __forceinline__ float gelu_exact(float x) {
    return 0.5f * x * (1.0f + erff(x * 0.70710678118654752440f));
}

// ---------------------------------------------------------------------------
// Kernel 1: 3-nearest-neighbor search. xyz2[b] staged in LDS (24 KB),
// one thread per query point keeps a register top-3 (by squared distance).
// ---------------------------------------------------------------------------
__global__ __launch_bounds__(256) void knn3_kernel(
    const float* __restrict__ xyz1, const float* __restrict__ xyz2,
    float* __restrict__ wbuf, int* __restrict__ ibuf)
{
    __shared__ float sx[S_], sy[S_], sz[S_];
    const int b = blockIdx.y;
    const float* p2 = xyz2 + (size_t)b * S_ * 3;
    for (int s = threadIdx.x; s < S_; s += 256) {
        sx[s] = p2[s * 3 + 0];
        sy[s] = p2[s * 3 + 1];
        sz[s] = p2[s * 3 + 2];
    }
    __syncthreads();

    const int n = blockIdx.x * 256 + threadIdx.x;
    const float* q = xyz1 + ((size_t)b * N_ + n) * 3;
    const float qx = q[0], qy = q[1], qz = q[2];

    float d0 = 1e30f, d1 = 1e30f, d2 = 1e30f;
    int   i0 = 0,     i1 = 0,     i2 = 0;
    for (int s = 0; s < S_; ++s) {
        const float dx = qx - sx[s];
        const float dy = qy - sy[s];
        const float dz = qz - sz[s];
        const float dd = dx * dx + dy * dy + dz * dz;
        if (dd < d2) {
            if (dd < d1) {
                d2 = d1; i2 = i1;
                if (dd < d0) { d1 = d0; i1 = i0; d0 = dd; i0 = s; }
                else         { d1 = dd; i1 = s; }
            } else { d2 = dd; i2 = s; }
        }
    }
    const float r0 = 1.0f / (sqrtf(fmaxf(d0, 0.0f)) + EPS_DIST);
    const float r1 = 1.0f / (sqrtf(fmaxf(d1, 0.0f)) + EPS_DIST);
    const float r2 = 1.0f / (sqrtf(fmaxf(d2, 0.0f)) + EPS_DIST);
    const float rs = 1.0f / (r0 + r1 + r2);
    const size_t o = ((size_t)b * N_ + n) * 3;
    wbuf[o + 0] = r0 * rs; wbuf[o + 1] = r1 * rs; wbuf[o + 2] = r2 * rs;
    ibuf[o + 0] = i0;      ibuf[o + 1] = i1;      ibuf[o + 2] = i2;
}

// ---------------------------------------------------------------------------
// Kernel 2: inverse-distance interpolation. Each block handles 512 points of
// one batch; per channel we stage the 2048-float points2 row in LDS and each
// thread gathers its 3 taps from LDS.
// ---------------------------------------------------------------------------
__global__ __launch_bounds__(256) void interp_kernel(
    const float* __restrict__ points2, const float* __restrict__ wbuf,
    const int* __restrict__ ibuf, float* __restrict__ interp)
{
    __shared__ float row[S_];
    const int b  = blockIdx.x;
    const int n0 = blockIdx.y * 512 + threadIdx.x;
    const int n1 = n0 + 256;

    const size_t o0 = ((size_t)b * N_ + n0) * 3;
    const size_t o1 = ((size_t)b * N_ + n1) * 3;
    const float w00 = wbuf[o0 + 0], w01 = wbuf[o0 + 1], w02 = wbuf[o0 + 2];
    const float w10 = wbuf[o1 + 0], w11 = wbuf[o1 + 1], w12 = wbuf[o1 + 2];
    const int   i00 = ibuf[o0 + 0], i01 = ibuf[o0 + 1], i02 = ibuf[o0 + 2];
    const int   i10 = ibuf[o1 + 0], i11 = ibuf[o1 + 1], i12 = ibuf[o1 + 2];

    const float* p2b  = points2 + (size_t)b * C2_ * S_;
    float*       outb = interp  + (size_t)b * C2_ * N_;

    for (int c = 0; c < C2_; ++c) {
        __syncthreads();
        const float* prow = p2b + (size_t)c * S_;
        for (int s = threadIdx.x; s < S_; s += 256) row[s] = prow[s];
        __syncthreads();
        outb[(size_t)c * N_ + n0] = w00 * row[i00] + w01 * row[i01] + w02 * row[i02];
        outb[(size_t)c * N_ + n1] = w10 * row[i10] + w11 * row[i11] + w12 * row[i12];
    }
}

// ---------------------------------------------------------------------------
// Kernel 3: time-embedding MLP  te[b][o] = GELU(t_embed[b]) @ lin_w^T + lin_b
// ---------------------------------------------------------------------------
__global__ __launch_bounds__(256) void temlp_kernel(
    const float* __restrict__ t_embed, const float* __restrict__ lin_w,
    const float* __restrict__ lin_b, float* __restrict__ te_out)
{
    __shared__ float g[TDIM_];
    const int b = blockIdx.x;
    if (threadIdx.x < TDIM_)
        g[threadIdx.x] = gelu_exact(t_embed[b * TDIM_ + threadIdx.x]);
    __syncthreads();
    const int o = threadIdx.x;
    const float* wr = lin_w + (size_t)o * TDIM_;
    float s = lin_b[o];
    for (int t = 0; t < TDIM_; ++t) s += g[t] * wr[t];
    te_out[b * M_ + o] = s;
}

// ---------------------------------------------------------------------------
// Staging helper: copy one K-tile (KT_ x NB_ of X, MB_ x KT_ of W) into LDS.
// Uses CDNA5 GLOBAL_LOAD_ASYNC_TO_LDS (ASYNCcnt-tracked) when available.
//   X tile: 16 rows x 256 cols -> 1024 b128 groups -> 4 per thread
//   W tile: 32 rows x 16  cols -> 256 b64 groups  -> 1 per thread
// ---------------------------------------------------------------------------
__device__ __forceinline__ void stage_tiles(
    float* __restrict__ xdst, float* __restrict__ wdst,
    const float* __restrict__ xsrc,   // points at [kt][col_base] of X source
    const float* __restrict__ wsrc,   // points at W[m0blk][kt]
    int K, int tid)
{
#if USE_ASYNC
    #pragma unroll
    for (int j = 0; j < 4; ++j) {
        const int g   = tid + j * 256;
        const int r   = g >> 6;
        const int c4  = (g & 63) << 2;
        __builtin_amdgcn_global_load_async_to_lds_b128(
            (AS1 i32x4*)(xsrc + (size_t)r * N_ + c4),
            (AS3 i32x4*)(xdst + r * NB_ + c4), 0, 0);
    }
    {
        const int r  = tid >> 3;
        const int c2 = (tid & 7) << 1;
        __builtin_amdgcn_global_load_async_to_lds_b64(
            (AS1 i32x2*)(wsrc + (size_t)r * K + c2),
            (AS3 i32x2*)(wdst + r * KT_ + c2), 0, 0);
    }
#else
    #pragma unroll
    for (int j = 0; j < 4; ++j) {
        const int g   = tid + j * 256;
        const int r   = g >> 6;
        const int c4  = (g & 63) << 2;
        *(float4*)(xdst + r * NB_ + c4) = *(const float4*)(xsrc + (size_t)r * N_ + c4);
    }
    {
        const int r  = tid >> 3;
        const int c2 = (tid & 7) << 1;
        *(v2f*)(wdst + r * KT_ + c2) = *(const v2f*)(wsrc + (size_t)r * K + c2);
    }
#endif
}

__device__ __forceinline__ void wait_async_lds() {
#if USE_ASYNC
    __builtin_amdgcn_s_wait_asynccnt(0);
#endif
}

// ---------------------------------------------------------------------------
// Kernel 4: WMMA f32 GEMM  H[b] = W(256xK) x X(KxN) + te[b][:,None]
// X rows [0,split) come from Xa, rows [split,K) from Xb (fused concat).
// Workgroup macro-tile: 32(M) x 256(N); 8 waves = 2(M) x 4(N), each wave a
// 16x64 strip (4 f32 accumulators). K-tiles of 16 double-buffered in LDS via
// async global->LDS copies; inner loop is ds_load + v_wmma_f32_16x16x4_f32.
// Fragment layouts per CDNA5 ISA 7.12.2:
//   A 16x4 : lane M = lane&15, VGPR pair = K (k+2*hi, k+2*hi+1)
//   B 4x16 : lane N = lane&15, lane half selects K half, VGPR pair = K pair
//   C 16x16: VGPR r -> row r + 8*hi, lane -> column
// ---------------------------------------------------------------------------
__global__ __launch_bounds__(256) void gemm_wmma_kernel(
    const float* __restrict__ W,
    const float* __restrict__ Xa, const float* __restrict__ Xb,
    long bstride_a, long bstride_b, int split, int K,
    const float* __restrict__ te, float* __restrict__ H)
{
    __shared__ float xb[2][KT_ * NB_];   // 2 x 16 x 256 floats = 32 KB
    __shared__ float wb[2][MB_ * KT_];   // 2 x 32 x 16  floats =  4 KB

    const int b     = blockIdx.z;
    const int m0blk = blockIdx.y * MB_;
    const int cb    = blockIdx.x * NB_;
    const int tid   = threadIdx.x;

    const int wave = tid >> 5;
    const int lane = tid & 31;
    const int hi   = lane >> 4;          // 0 | 1
    const int lx   = lane & 15;
    const int wm   = wave & 1;           // M sub-block (0..1)
    const int wn   = wave >> 1;          // N sub-block (0..3)
    const int mrow = wm * 16 + lx;       // local A row for this lane
    const int nl0  = wn * 64 + lx;       // local col base for this lane

    const v8f zero = {0.f, 0.f, 0.f, 0.f, 0.f, 0.f, 0.f, 0.f};
    v8f acc[4] = {zero, zero, zero, zero};

    // Source row pointer for a K-tile (split is a multiple of KT_)
    const float* aBase = Xa + (size_t)b * bstride_a;
    const float* bBase = Xb + (size_t)b * bstride_b;

    // Prologue: stage first K-tile into buffer 0
    {
        const float* xs = (0 < split) ? aBase : bBase;
        stage_tiles(xb[0], wb[0], xs + cb, W + (size_t)m0blk * K, K, tid);
    }

    int cur = 0;
    for (int kt = 0; kt < K; kt += KT_) {
        wait_async_lds();
        __syncthreads();

        const int ktn = kt + KT_;
        if (ktn < K) {
            const float* xs = (ktn < split)
                ? (aBase + (size_t)ktn * N_)
                : (bBase + (size_t)(ktn - split) * N_);
            stage_tiles(xb[cur ^ 1], wb[cur ^ 1], xs + cb,
                        W + (size_t)m0blk * K + ktn, K, tid);
        }

        #pragma unroll
        for (int kk = 0; kk < KT_; kk += 4) {
            const v2f a = *(const v2f*)&wb[cur][mrow * KT_ + kk + 2 * hi];
            #pragma unroll
            for (int t = 0; t < 4; ++t) {
                const int nl = nl0 + t * 16;
                v2f bf;
                bf.x = xb[cur][(kk + 2 * hi)     * NB_ + nl];
                bf.y = xb[cur][(kk + 2 * hi + 1) * NB_ + nl];
                acc[t] = __builtin_amdgcn_wmma_f32_16x16x4_f32(
                    false, a, false, bf, (short)0, acc[t], false, false);
            }
        }
        __syncthreads();
        cur ^= 1;
    }

    // Epilogue: add per-channel time-embedding bias, store pre-BN activations.
    const int m0 = m0blk + wm * 16;
    const float* teb = te + b * M_;
    float* Hb = H + ((size_t)b * M_ + m0) * N_;
    #pragma unroll
    for (int t = 0; t < 4; ++t) {
        const int n = cb + nl0 + t * 16;
        #pragma unroll
        for (int r = 0; r < 8; ++r) {
            const int m = r + 8 * hi;
            Hb[(size_t)m * N_ + n] = acc[t][r] + teb[m0 + m];
        }
    }
}

// ---------------------------------------------------------------------------
// Kernel 5: BatchNorm batch statistics, one block per channel (reduce B*N).
// ---------------------------------------------------------------------------
__global__ __launch_bounds__(256) void bnstats_kernel(
    const float* __restrict__ H, float* __restrict__ mean, float* __restrict__ var)
{
    const int c = blockIdx.x;
    float s = 0.f, s2 = 0.f;
    for (int b = 0; b < B_; ++b) {
        const float* p = H + ((size_t)b * M_ + c) * N_;
        for (int n = threadIdx.x; n < N_; n += 256) {
            const float v = p[n];
            s += v; s2 += v * v;
        }
    }
    __shared__ float rs[256], rq[256];
    rs[threadIdx.x] = s; rq[threadIdx.x] = s2;
    __syncthreads();
    for (int off = 128; off > 0; off >>= 1) {
        if (threadIdx.x < off) {
            rs[threadIdx.x] += rs[threadIdx.x + off];
            rq[threadIdx.x] += rq[threadIdx.x + off];
        }
        __syncthreads();
    }
    if (threadIdx.x == 0) {
        const float inv = 1.0f / (float)(B_ * N_);
        const float m = rs[0] * inv;
        mean[c] = m;
        var[c]  = rq[0] * inv - m * m;
    }
}

// ---------------------------------------------------------------------------
// Kernel 6: normalize + affine + exact GELU (elementwise).
// ---------------------------------------------------------------------------
__global__ __launch_bounds__(256) void bnact_kernel(
    const float* __restrict__ H, const float* __restrict__ mean,
    const float* __restrict__ var, const float* __restrict__ g,
    const float* __restrict__ bb, float* __restrict__ out)
{
    const size_t i = (size_t)blockIdx.x * 256 + threadIdx.x;
    const int c = (int)((i / N_) % M_);
    const float x = H[i];
    const float y = (x - mean[c]) * rsqrtf(var[c] + EPS_BN) * g[c] + bb[c];
    out[i] = gelu_exact(y);
}

// ---------------------------------------------------------------------------
extern "C" void kernel_launch(void* const* d_in, const int* in_sizes, int n_in,
                              void* d_out, int out_size, void* d_ws, size_t ws_size,
                              hipStream_t stream) {
    (void)in_sizes; (void)n_in; (void)out_size; (void)ws_size;
    const float* xyz1    = (const float*)d_in[0];
    const float* xyz2    = (const float*)d_in[1];
    const float* points1 = (const float*)d_in[2];
    const float* points2 = (const float*)d_in[3];
    const float* t_embed = (const float*)d_in[4];
    const float* conv_w0 = (const float*)d_in[5];
    const float* lin_w0  = (const float*)d_in[6];
    const float* lin_b0  = (const float*)d_in[7];
    const float* bn_g0   = (const float*)d_in[8];
    const float* bn_b0   = (const float*)d_in[9];
    const float* conv_w1 = (const float*)d_in[10];
    const float* lin_w1  = (const float*)d_in[11];
    const float* lin_b1  = (const float*)d_in[12];
    const float* bn_g1   = (const float*)d_in[13];
    const float* bn_b1   = (const float*)d_in[14];
    float* out = (float*)d_out;

    // Workspace carve-up (floats)
    float* ws     = (float*)d_ws;
    const size_t nKnn  = (size_t)B_ * N_ * 3;          // 196608
    const size_t nFeat = (size_t)B_ * C2_ * N_;        // 16777216
    float* wbuf   = ws;
    int*   ibuf   = (int*)(ws + nKnn);
    float* interp = ws + 2 * nKnn;                     // reused as X1 after layer 0
    float* H      = interp + nFeat;
    float* te0    = H + nFeat;
    float* te1    = te0 + B_ * M_;
    float* meanv  = te1 + B_ * M_;
    float* varv   = meanv + M_;

    // 1) 3-NN weights/indices
    knn3_kernel<<<dim3(N_ / 256, B_), 256, 0, stream>>>(xyz1, xyz2, wbuf, ibuf);

    // 2) feature interpolation
    interp_kernel<<<dim3(B_, N_ / 512), 256, 0, stream>>>(points2, wbuf, ibuf, interp);

    // 3) time-embedding MLPs
    temlp_kernel<<<dim3(B_), 256, 0, stream>>>(t_embed, lin_w0, lin_b0, te0);
    temlp_kernel<<<dim3(B_), 256, 0, stream>>>(t_embed, lin_w1, lin_b1, te1);

    // 4) layer 0 GEMM: [256 x 384] x [384 x N], rows 0..127 from points1,
    //    rows 128..383 from interp (fused concat)
    gemm_wmma_kernel<<<dim3(N_ / NB_, M_ / MB_, B_), 256, 0, stream>>>(
        conv_w0, points1, interp, (long)C1_ * N_, (long)C2_ * N_, C1_, K0_, te0, H);

    // 5) BN stats + normalize/GELU -> X1 (reuses interp buffer)
    bnstats_kernel<<<dim3(M_), 256, 0, stream>>>(H, meanv, varv);
    bnact_kernel<<<dim3((unsigned)((size_t)B_ * M_ * N_ / 256)), 256, 0, stream>>>(
        H, meanv, varv, bn_g0, bn_b0, interp);

    // 6) layer 1 GEMM: [256 x 256] x [256 x N]
    gemm_wmma_kernel<<<dim3(N_ / NB_, M_ / MB_, B_), 256, 0, stream>>>(
        conv_w1, interp, interp, (long)C2_ * N_, (long)C2_ * N_, C2_, C2_, te1, H);

    // 7) BN stats + normalize/GELU -> d_out
    bnstats_kernel<<<dim3(M_), 256, 0, stream>>>(H, meanv, varv);
    bnact_kernel<<<dim3((unsigned)((size_t)B_ * M_ * N_ / 256)), 256, 0, stream>>>(
        H, meanv, varv, bn_g1, bn_b1, out);
}